// ShapeAwarePhaseAwareLoss_74131135529095
// MI455X (gfx1250) — compile-verified
//
#include <hip/hip_runtime.h>
#include <hip/hip_bf16.h>

// ---------------------------------------------------------------------------
// ShapeAwarePhaseAwareLoss for MI455X (gfx1250, wave32)
//
// out = 2 * mean_b[ (mean_d((sp-st)^2) + mean_d((cp-ct)^2)) * factor(b) ]
//
// HBM-bound (1 GiB streamed once -> ~46us floor @ 23.3 TB/s). Strategy:
//   * b128 non-temporal loads (stream, bypass cache pollution; footprint >> L2)
//   * per-thread f32 FMA partials
//   * per-wave cross-lane reduction via V_WMMA_F32_16X16X4_F32 (exact f32)
//   * deterministic two-stage tree reduction through d_ws (no atomics)
// ---------------------------------------------------------------------------

typedef __attribute__((ext_vector_type(4))) float f4;
typedef __attribute__((ext_vector_type(2))) float v2f;
typedef __attribute__((ext_vector_type(8))) float v8f;

// Exact f32 wave32 reduction using the matrix pipe.
// A (16x4 f32, 2 VGPRs): VGPR0 = acc -> A[m][0]=acc[m] (lanes 0-15),
// A[m][2]=acc[m+16] (lanes 16-31); VGPR1 = 0. B = all ones.
// D[m][n] = acc[m] + acc[m+16] for every n. Summing the 8 C VGPRs gives,
// in lane n (<16): sum_{m=0..7} D[m][n], in lane n+16: sum_{m=8..15} D[m][n].
// One xor-16 shuffle add -> every lane holds the full 32-lane total.
__device__ __forceinline__ float wave_reduce_f32_wmma(float acc) {
    v2f a; a.x = acc;  a.y = 0.0f;
    v2f b; b.x = 1.0f; b.y = 1.0f;
    v8f c = {};
    c = __builtin_amdgcn_wmma_f32_16x16x4_f32(
        /*neg_a=*/false, a, /*neg_b=*/false, b,
        /*c_mod=*/(short)0, c, /*reuse_a=*/false, /*reuse_b=*/false);
    float s = ((c[0] + c[1]) + (c[2] + c[3])) + ((c[4] + c[5]) + (c[6] + c[7]));
    s += __shfl_xor(s, 16, 32);
    return s;
}

// One block per batch row. Streams both (sin,cos) pred/true pairs for the row,
// accumulates sum of squared diffs, applies the per-row shape factor, writes
// one weighted partial per row into ws[row].
__global__ __launch_bounds__(256) void loss_stage1(
    const float* __restrict__ sp, const float* __restrict__ cp,
    const float* __restrict__ st, const float* __restrict__ ct,
    const float* __restrict__ shape_types,
    float* __restrict__ ws, int nvec4 /* = D/4 */) {
    const int row = blockIdx.x;
    const int tid = threadIdx.x;
    const long long base = (long long)row * (long long)(nvec4 * 4);

    const f4* __restrict__ sp4 = (const f4*)(sp + base);
    const f4* __restrict__ st4 = (const f4*)(st + base);
    const f4* __restrict__ cp4 = (const f4*)(cp + base);
    const f4* __restrict__ ct4 = (const f4*)(ct + base);

    float acc0 = 0.0f, acc1 = 0.0f;
    #pragma unroll 4
    for (int i = tid; i < nvec4; i += 256) {
        const f4 a = __builtin_nontemporal_load(&sp4[i]);   // global_load_b128 th:NT
        const f4 b = __builtin_nontemporal_load(&st4[i]);
        const f4 p = __builtin_nontemporal_load(&cp4[i]);
        const f4 q = __builtin_nontemporal_load(&ct4[i]);
        float d0 = a.x - b.x, d1 = a.y - b.y, d2 = a.z - b.z, d3 = a.w - b.w;
        float e0 = p.x - q.x, e1 = p.y - q.y, e2 = p.z - q.z, e3 = p.w - q.w;
        acc0 = fmaf(d0, d0, acc0); acc1 = fmaf(d1, d1, acc1);
        acc0 = fmaf(d2, d2, acc0); acc1 = fmaf(d3, d3, acc1);
        acc0 = fmaf(e0, e0, acc0); acc1 = fmaf(e1, e1, acc1);
        acc0 = fmaf(e2, e2, acc0); acc1 = fmaf(e3, e3, acc1);
    }

    // Uniform control flow, EXEC all-ones: safe for WMMA.
    const float wtot = wave_reduce_f32_wmma(acc0 + acc1);

    __shared__ float smem[8];
    const int wave = tid >> 5;
    const int lane = tid & 31;
    if (lane == 0) smem[wave] = wtot;
    __syncthreads();
    if (tid == 0) {
        float s = 0.0f;
        #pragma unroll
        for (int w = 0; w < 8; ++w) s += smem[w];
        const float factor = (shape_types[row] < 0.5f) ? 1.0f : 1.3f;
        ws[row] = s * factor;
    }
}

// Fixed-tree reduction of the B per-row weighted partials; applies 2/(B*D).
__global__ __launch_bounds__(1024) void loss_stage2(
    const float* __restrict__ ws, float* __restrict__ out,
    int nrows, float scale) {
    __shared__ float smem[1024];
    const int tid = threadIdx.x;
    float s = 0.0f;
    for (int i = tid; i < nrows; i += 1024) s += ws[i];
    smem[tid] = s;
    __syncthreads();
    #pragma unroll
    for (int stride = 512; stride > 0; stride >>= 1) {
        if (tid < stride) smem[tid] += smem[tid + stride];
        __syncthreads();
    }
    if (tid == 0) out[0] = smem[0] * scale;
}

extern "C" void kernel_launch(void* const* d_in, const int* in_sizes, int n_in,
                              void* d_out, int out_size, void* d_ws, size_t ws_size,
                              hipStream_t stream) {
    // setup_inputs order: phase_sin_pred, phase_cos_pred, phase_sin_true,
    //                     phase_cos_true, shape_types
    const float* sp    = (const float*)d_in[0];
    const float* cp    = (const float*)d_in[1];
    const float* st    = (const float*)d_in[2];
    const float* ct    = (const float*)d_in[3];
    const float* shape = (const float*)d_in[4];

    const int B = in_sizes[4];                 // 4096
    const int D = in_sizes[0] / B;             // 16384
    const int nvec4 = D / 4;

    float* ws  = (float*)d_ws;                 // B floats of scratch
    float* out = (float*)d_out;

    const float scale = 2.0f / ((float)B * (float)D); // exact: 2^-25 here

    loss_stage1<<<B, 256, 0, stream>>>(sp, cp, st, ct, shape, ws, nvec4);
    loss_stage2<<<1, 1024, 0, stream>>>(ws, out, B, scale);
}